// ReLinkGNN_37443524886863
// MI455X (gfx1250) — compile-verified
//
#include <hip/hip_runtime.h>
#include <math.h>

#define N_NODES 50000
#define N_EDGES 800000
#define N_TOT   (N_EDGES + N_NODES)   // edges + self loops
#define IN_CH   64
#define HID_CH  128
#define OUT_CH  3

typedef __attribute__((ext_vector_type(2))) float v2f;
typedef __attribute__((ext_vector_type(8))) float v8f;

// ---------------------------------------------------------------------------
// Workspace layout (floats):
//   deg/dinv : [50048]        (deg computed in place, then rsqrt'd)
//   h1       : [6,400,000]    X @ W1          (50000 x 128)
//   agg1     : [6,400,000]    scatter target  (50000 x 128)
//   h2       : [150016]       relu(agg1+b1) @ W2  (50000 x 3)
//   agg2     : [150000]       scatter target  (50000 x 3)
// total ~52.6 MB
// ---------------------------------------------------------------------------
#define WS_DEG  0
#define WS_H1   50048
#define WS_AGG1 (WS_H1 + (size_t)N_NODES * HID_CH)
#define WS_H2   (WS_AGG1 + (size_t)N_NODES * HID_CH)
#define WS_AGG2 (WS_H2 + 150016)

// ---- init: zero agg1/agg2, deg = 1.0 (self loop) ---------------------------
__global__ __launch_bounds__(256) void k_init(float* __restrict__ agg1,
                                              float* __restrict__ agg2,
                                              float* __restrict__ deg) {
  const int i = blockIdx.x * blockDim.x + threadIdx.x;
  if (i < N_NODES * HID_CH) agg1[i] = 0.0f;
  if (i < N_NODES * OUT_CH) agg2[i] = 0.0f;
  if (i < N_NODES)          deg[i]  = 1.0f;   // self-loop contribution
}

// ---- degree: deg[col[e]] += 1 ---------------------------------------------
__global__ __launch_bounds__(256) void k_deg(const int* __restrict__ col,
                                             float* __restrict__ deg) {
  const int e = blockIdx.x * blockDim.x + threadIdx.x;
  if (e < N_EDGES) atomicAdd(&deg[col[e]], 1.0f);
}

// ---- dinv = rsqrt(deg) (deg >= 1 always, keep guard for parity) -----------
__global__ __launch_bounds__(256) void k_dinv(float* __restrict__ deg) {
  const int i = blockIdx.x * blockDim.x + threadIdx.x;
  if (i < N_NODES) {
    const float d = deg[i];
    deg[i] = (d > 0.0f) ? rsqrtf(d) : 0.0f;
  }
}

// ---- GEMM1 via V_WMMA_F32_16X16X4_F32: h1 = x @ W1 ------------------------
// Block: 256 threads = 8 waves. Each wave computes a 16(M) x 128(N) tile.
// W1 (64x128 = 32 KB) staged in LDS once per block.
__global__ __launch_bounds__(256) void k_gemm1(const float* __restrict__ x,
                                               const float* __restrict__ W1,
                                               float* __restrict__ h1) {
  __shared__ float lW[IN_CH * HID_CH];   // 32 KB of the 320 KB/WGP LDS
  const int tid = threadIdx.x;

  // cooperative 128-bit copies of W1 into LDS
  for (int i = tid * 4; i < IN_CH * HID_CH; i += 256 * 4) {
    *(float4*)(&lW[i]) = *(const float4*)(&W1[i]);
  }
  __syncthreads();

  const int wave = tid >> 5;
  const int lane = tid & 31;
  const int half = lane >> 4;   // selects K pair (A/B) and M+8 (C/D)
  const int l    = lane & 15;

  const int m0 = blockIdx.x * 128 + wave * 16;

  // A-fragment row for this lane (clamped so EXEC stays all-1s for WMMA)
  int arow = m0 + l;
  if (arow > N_NODES - 1) arow = N_NODES - 1;
  const float* aptr = x + (size_t)arow * IN_CH;

  v8f acc[8] = {};   // 8 N-tiles of 16 -> full 128 columns

#pragma unroll
  for (int kt = 0; kt < IN_CH / 4; ++kt) {       // 16 K-steps of K=4
    const int kbase = kt * 4 + 2 * half;
    v2f a;
    a.x = aptr[kbase + 0];
    a.y = aptr[kbase + 1];
#pragma unroll
    for (int nt = 0; nt < 8; ++nt) {
      v2f b;
      b.x = lW[(kbase + 0) * HID_CH + nt * 16 + l];
      b.y = lW[(kbase + 1) * HID_CH + nt * 16 + l];
      acc[nt] = __builtin_amdgcn_wmma_f32_16x16x4_f32(
          /*neg_a=*/false, a, /*neg_b=*/false, b,
          /*c_mod=*/(short)0, acc[nt], /*reuse_a=*/false, /*reuse_b=*/false);
    }
  }

  // Store D: VGPR r holds M = r (lanes 0-15) / r+8 (lanes 16-31), N = l
#pragma unroll
  for (int nt = 0; nt < 8; ++nt) {
#pragma unroll
    for (int r = 0; r < 8; ++r) {
      const int mrow = m0 + r + 8 * half;
      if (mrow < N_NODES) h1[(size_t)mrow * HID_CH + nt * 16 + l] = acc[nt][r];
    }
  }
}

// ---- scatter layer 1: one wave per edge, float4 per lane ------------------
__global__ __launch_bounds__(256) void k_scatter1(const int* __restrict__ row,
                                                  const int* __restrict__ col,
                                                  const float* __restrict__ dinv,
                                                  const float* __restrict__ h1,
                                                  float* __restrict__ agg1) {
  const long e = (long)((blockIdx.x * blockDim.x + threadIdx.x) >> 5);
  const int lane = threadIdx.x & 31;
  if (e >= N_TOT) return;                  // uniform per wave
  int r, c;
  if (e < N_EDGES) { r = row[e]; c = col[e]; }
  else             { r = c = (int)(e - N_EDGES); }   // self loop
  const float w = dinv[r] * dinv[c];

  const float4 v = *(const float4*)(h1 + (size_t)r * HID_CH + lane * 4);
  float* dst = agg1 + (size_t)c * HID_CH + lane * 4;
  atomicAdd(dst + 0, v.x * w);
  atomicAdd(dst + 1, v.y * w);
  atomicAdd(dst + 2, v.z * w);
  atomicAdd(dst + 3, v.w * w);
}

// ---- fused bias + relu + GEMM2: one wave per node -------------------------
__global__ __launch_bounds__(256) void k_layer2(const float* __restrict__ agg1,
                                                const float* __restrict__ b1,
                                                const float* __restrict__ W2,
                                                float* __restrict__ h2) {
  const int node = (blockIdx.x * blockDim.x + threadIdx.x) >> 5;
  const int lane = threadIdx.x & 31;
  if (node >= N_NODES) return;

  float4 v  = *(const float4*)(agg1 + (size_t)node * HID_CH + lane * 4);
  const float4 bb = *(const float4*)(b1 + lane * 4);
  v.x = fmaxf(v.x + bb.x, 0.0f);
  v.y = fmaxf(v.y + bb.y, 0.0f);
  v.z = fmaxf(v.z + bb.z, 0.0f);
  v.w = fmaxf(v.w + bb.w, 0.0f);

  const int ch = lane * 4;
  float p0 = v.x * W2[(ch + 0) * OUT_CH + 0] + v.y * W2[(ch + 1) * OUT_CH + 0] +
             v.z * W2[(ch + 2) * OUT_CH + 0] + v.w * W2[(ch + 3) * OUT_CH + 0];
  float p1 = v.x * W2[(ch + 0) * OUT_CH + 1] + v.y * W2[(ch + 1) * OUT_CH + 1] +
             v.z * W2[(ch + 2) * OUT_CH + 1] + v.w * W2[(ch + 3) * OUT_CH + 1];
  float p2 = v.x * W2[(ch + 0) * OUT_CH + 2] + v.y * W2[(ch + 1) * OUT_CH + 2] +
             v.z * W2[(ch + 2) * OUT_CH + 2] + v.w * W2[(ch + 3) * OUT_CH + 2];

  // wave32 butterfly reduction
#pragma unroll
  for (int m = 16; m >= 1; m >>= 1) {
    p0 += __shfl_xor(p0, m, 32);
    p1 += __shfl_xor(p1, m, 32);
    p2 += __shfl_xor(p2, m, 32);
  }
  if (lane == 0) {
    float* o = h2 + (size_t)node * OUT_CH;
    o[0] = p0; o[1] = p1; o[2] = p2;
  }
}

// ---- scatter layer 2: one thread per edge, 3 channels ---------------------
__global__ __launch_bounds__(256) void k_scatter2(const int* __restrict__ row,
                                                  const int* __restrict__ col,
                                                  const float* __restrict__ dinv,
                                                  const float* __restrict__ h2,
                                                  float* __restrict__ agg2) {
  const int e = blockIdx.x * blockDim.x + threadIdx.x;
  if (e >= N_TOT) return;
  int r, c;
  if (e < N_EDGES) { r = row[e]; c = col[e]; }
  else             { r = c = e - N_EDGES; }
  const float w = dinv[r] * dinv[c];
  const float* hp = h2 + (size_t)r * OUT_CH;
  float* dst = agg2 + (size_t)c * OUT_CH;
  atomicAdd(dst + 0, hp[0] * w);
  atomicAdd(dst + 1, hp[1] * w);
  atomicAdd(dst + 2, hp[2] * w);
}

// ---- bias + log_softmax ---------------------------------------------------
__global__ __launch_bounds__(256) void k_final(const float* __restrict__ agg2,
                                               const float* __restrict__ b2,
                                               float* __restrict__ out) {
  const int i = blockIdx.x * blockDim.x + threadIdx.x;
  if (i >= N_NODES) return;
  const float v0 = agg2[i * 3 + 0] + b2[0];
  const float v1 = agg2[i * 3 + 1] + b2[1];
  const float v2 = agg2[i * 3 + 2] + b2[2];
  const float m  = fmaxf(v0, fmaxf(v1, v2));
  const float s  = __expf(v0 - m) + __expf(v1 - m) + __expf(v2 - m);
  const float lse = m + __logf(s);
  out[i * 3 + 0] = v0 - lse;
  out[i * 3 + 1] = v1 - lse;
  out[i * 3 + 2] = v2 - lse;
}

extern "C" void kernel_launch(void* const* d_in, const int* in_sizes, int n_in,
                              void* d_out, int out_size, void* d_ws, size_t ws_size,
                              hipStream_t stream) {
  (void)in_sizes; (void)n_in; (void)out_size; (void)ws_size;

  const float* x   = (const float*)d_in[0];            // [50000, 64]
  const int*   ei  = (const int*)d_in[1];              // [2, 800000]
  const float* W1  = (const float*)d_in[2];            // [64, 128]
  const float* b1  = (const float*)d_in[3];            // [128]
  const float* W2  = (const float*)d_in[4];            // [128, 3]
  const float* b2  = (const float*)d_in[5];            // [3]
  float*       out = (float*)d_out;                    // [50000, 3]

  const int* row = ei;                // edge_index[0] : sources
  const int* col = ei + N_EDGES;      // edge_index[1] : targets

  float* ws   = (float*)d_ws;
  float* deg  = ws + WS_DEG;          // becomes dinv in place
  float* h1   = ws + WS_H1;
  float* agg1 = ws + WS_AGG1;
  float* h2   = ws + WS_H2;
  float* agg2 = ws + WS_AGG2;

  const int B = 256;
  // 1) init accumulators + self-loop degree
  {
    const int n = N_NODES * HID_CH;
    k_init<<<(n + B - 1) / B, B, 0, stream>>>(agg1, agg2, deg);
  }
  // 2) degree atomics over real edges
  k_deg<<<(N_EDGES + B - 1) / B, B, 0, stream>>>(col, deg);
  // 3) dinv = rsqrt(deg)
  k_dinv<<<(N_NODES + B - 1) / B, B, 0, stream>>>(deg);
  // 4) h1 = x @ W1  (WMMA f32 16x16x4, 128 rows per block)
  k_gemm1<<<(N_NODES + 127) / 128, B, 0, stream>>>(x, W1, h1);
  // 5) normalized scatter-add, layer 1 (8 edges / 256-thread block)
  {
    const long waves = N_TOT;
    k_scatter1<<<(unsigned)((waves + 7) / 8), B, 0, stream>>>(row, col, deg, h1, agg1);
  }
  // 6) bias + relu + GEMM2 fused (8 nodes / block)
  k_layer2<<<(N_NODES + 7) / 8, B, 0, stream>>>(agg1, b1, W2, h2);
  // 7) normalized scatter-add, layer 2
  k_scatter2<<<(N_TOT + B - 1) / B, B, 0, stream>>>(row, col, deg, h2, agg2);
  // 8) bias + log_softmax
  k_final<<<(N_NODES + B - 1) / B, B, 0, stream>>>(agg2, b2, out);
}